// NaiveAsymmetricCorrelationBlock_64527588655459
// MI455X (gfx1250) — compile-verified
//
#include <hip/hip_runtime.h>

// Asymmetric correlation (cost volume) via banded-Gram WMMA (f32 16x16x4),
// double-buffered LDS chunks fed by GLOBAL_LOAD_ASYNC_TO_LDS_B128 (ASYNCcnt).
// out[n, (dh+4)*9+(dw+4), y, x] = (1/256) * sum_c x1[n,c,y,x] * x2[n,c,y+dh,x+dw]
// Grid: (9 dh, 96 y, 8 n). Block: 320 threads = 10 wave32, one 16-wide x-tile per wave.

typedef __attribute__((ext_vector_type(2))) float v2f;
typedef __attribute__((ext_vector_type(8))) float v8f;
typedef __attribute__((ext_vector_type(4))) int   v4i;

#define N_BATCH 8
#define C_CH    256
#define H_DIM   96
#define W_DIM   160
#define HW      (H_DIM * W_DIM)
#define DISP    4
#define NDISP   9
#define KC      16      // channels staged per LDS chunk
#define NCHUNK  (C_CH / KC)
#define X1W     168     // LDS row stride (dwords), 2*168 mod 64 = 16 -> conflict-free half-split
#define X2W     184     // LDS row stride (dwords), 2*184 mod 64 = 48 -> conflict-free
#define NTHREADS 320

#ifndef __has_builtin
#define __has_builtin(x) 0
#endif

#if __has_builtin(__builtin_amdgcn_global_load_async_to_lds_b128)
#define HAVE_ASYNC_LDS 1
#else
#define HAVE_ASYNC_LDS 0
#endif

#if HAVE_ASYNC_LDS
typedef __attribute__((address_space(1))) v4i gv4i;  // global (AS1) v4i
typedef __attribute__((address_space(3))) v4i lv4i;  // LDS (AS3) v4i
static __device__ __forceinline__ void async_ld16(const float* g, float* l) {
    // global -> LDS DMA, 16 bytes, tracked with ASYNCcnt.
    // AS1 ptr from flat address (identical representation); AS3 ptr = low 32 bits
    // of the generic LDS address.
    __builtin_amdgcn_global_load_async_to_lds_b128(
        (gv4i*)(uintptr_t)g, (lv4i*)(unsigned int)(uintptr_t)l, 0, 0);
}
#define ASYNC_AWAIT() asm volatile("s_wait_asynccnt 0x0" ::: "memory")
#else
static __device__ __forceinline__ void async_ld16(const float* g, float* l) {
    *(float4*)l = *(const float4*)g;
}
#define ASYNC_AWAIT() ((void)0)
#endif

__global__ __launch_bounds__(NTHREADS)
void corr_wmma_f32_kernel(const float* __restrict__ x1,
                          const float* __restrict__ x2,
                          float* __restrict__ out)
{
    const int dhi = blockIdx.x;   // 0..8 -> dh = dhi - 4
    const int y   = blockIdx.y;   // 0..95
    const int n   = blockIdx.z;   // 0..7

    const int tid  = threadIdx.x;
    const int lane = tid & 31;
    const int wv   = tid >> 5;    // 0..9 : x-tile index
    const int x0   = wv * 16;

    float* outp = out + ((size_t)n * (NDISP * NDISP) + (size_t)dhi * NDISP) * (size_t)HW
                      + (size_t)y * W_DIM;

    const int y2 = y + dhi - DISP;
    if (y2 < 0 || y2 >= H_DIM) {
        for (int i = tid; i < NDISP * W_DIM; i += NTHREADS) {
            int d = i / W_DIM, x = i - d * W_DIM;
            outp[(size_t)d * HW + x] = 0.0f;
        }
        return;
    }

    __shared__ float x1s[2][KC][X1W];   // x1[y] row tile, KC channels x 160, double-buffered
    __shared__ float x2s[2][KC][X2W];   // x2[y+dh] window (idx = pos+4), double-buffered
    __shared__ float obuf[NDISP][W_DIM];

    const float* x1row = x1 + (((size_t)n * C_CH) * H_DIM + (size_t)y ) * W_DIM;
    const float* x2row = x2 + (((size_t)n * C_CH) * H_DIM + (size_t)y2) * W_DIM;

    // Window pads (pos < 0 or >= 160 -> idx 0..3 and 164..175) are chunk-invariant: zero once.
    for (int i = tid; i < 2 * KC * 16; i += NTHREADS) {
        int b = i >> 8;              // buffer
        int r = (i >> 4) & (KC - 1); // k row
        int p = i & 15;              // pad slot
        x2s[b][r][(p < 4) ? p : (160 + p)] = 0.0f;
    }

    // Fixed per-thread staging assignment: 320 threads = 8 k-rows x 40 float4 columns.
    const int sx  = (tid % 40) * 4;  // x offset (floats), 16B aligned
    const int skb = tid / 40;        // 0..7 ; rows skb and skb+8 per chunk

    v8f acc0 = {};   // Gram block, columns x0-4 .. x0+11
    v8f acc1 = {};   // Gram block, columns x0+12 .. x0+27
    const int m = lane & 15;    // N (and A-row M index within half)
    const int h = lane >> 4;    // lane-half selector

    // Stage chunk 0 into buffer 0.
    {
        const float* g1 = x1row + (size_t)skb * HW + sx;
        const float* g2 = x2row + (size_t)skb * HW + sx;
        async_ld16(g1,            &x1s[0][skb    ][sx]);
        async_ld16(g1 + 8 * HW,   &x1s[0][skb + 8][sx]);
        async_ld16(g2,            &x2s[0][skb    ][4 + sx]);
        async_ld16(g2 + 8 * HW,   &x2s[0][skb + 8][4 + sx]);
    }
    ASYNC_AWAIT();
    __syncthreads();

    for (int c = 0; c < NCHUNK; ++c) {
        const int buf = c & 1;
        // Prefetch chunk c+1 into the other buffer (DMA overlapped with WMMA).
        if (c + 1 < NCHUNK) {
            const size_t kofs = (size_t)((c + 1) * KC + skb) * HW + sx;
            const float* g1 = x1row + kofs;
            const float* g2 = x2row + kofs;
            async_ld16(g1,            &x1s[buf ^ 1][skb    ][sx]);
            async_ld16(g1 + 8 * HW,   &x1s[buf ^ 1][skb + 8][sx]);
            async_ld16(g2,            &x2s[buf ^ 1][skb    ][4 + sx]);
            async_ld16(g2 + 8 * HW,   &x2s[buf ^ 1][skb + 8][4 + sx]);
        }

        // 16x16x4 fp32 WMMA: A = x1 tile (M=x, K=c), B = x2 window (K=c, N=col).
        // ISA layout: operand VGPR j on lane-half h holds K = 2*h + j.
#pragma unroll
        for (int kk = 0; kk < KC; kk += 4) {
            v2f a, b0, b1;
            a.x  = x1s[buf][kk + 2 * h + 0][x0 + m];
            a.y  = x1s[buf][kk + 2 * h + 1][x0 + m];
            b0.x = x2s[buf][kk + 2 * h + 0][x0 + m];       // window idx x0+N    -> pos x0+N-4
            b0.y = x2s[buf][kk + 2 * h + 1][x0 + m];
            b1.x = x2s[buf][kk + 2 * h + 0][x0 + 16 + m];  // window idx x0+16+N -> pos x0+N+12
            b1.y = x2s[buf][kk + 2 * h + 1][x0 + 16 + m];
            acc0 = __builtin_amdgcn_wmma_f32_16x16x4_f32(false, a, false, b0, (short)0, acc0, false, false);
            acc1 = __builtin_amdgcn_wmma_f32_16x16x4_f32(false, a, false, b1, (short)0, acc1, false, false);
        }

        ASYNC_AWAIT();       // wave's async loads into buf^1 have landed in LDS
        __syncthreads();     // all waves done reading buf / writing buf^1
    }

    // Extract the 9 band diagonals: D layout lane->N=m, vgpr r -> M=r+8*h.
    // dw+4 = 16*t + N - M ; each (M, dw) pair is covered by exactly one (t, N).
    const float scale = 1.0f / (float)C_CH;
#pragma unroll
    for (int r = 0; r < 8; ++r) {
        int M = r + 8 * h;
        int b0i = m - M;            // t = 0
        if (b0i >= 0 && b0i <= 8)
            obuf[b0i][x0 + M] = acc0[r] * scale;
        int b1i = 16 + m - M;       // t = 1
        if (b1i >= 0 && b1i <= 8)
            obuf[b1i][x0 + M] = acc1[r] * scale;
    }
    __syncthreads();

    // Coalesced store of the 9 displacement rows for this (n, y, dh).
    for (int i = tid; i < NDISP * W_DIM; i += NTHREADS) {
        int d = i / W_DIM, x = i - d * W_DIM;
        outp[(size_t)d * HW + x] = obuf[d][x];
    }
}

extern "C" void kernel_launch(void* const* d_in, const int* in_sizes, int n_in,
                              void* d_out, int out_size, void* d_ws, size_t ws_size,
                              hipStream_t stream) {
    const float* x1 = (const float*)d_in[0];
    const float* x2 = (const float*)d_in[1];
    float* out = (float*)d_out;
    dim3 grid(NDISP, H_DIM, N_BATCH);   // dh fastest -> 9 blocks share one x1 row in L2
    corr_wmma_f32_kernel<<<grid, NTHREADS, 0, stream>>>(x1, x2, out);
}